// FraudGNN_463856468591
// MI455X (gfx1250) — compile-verified
//
#include <hip/hip_runtime.h>
#include <math.h>

// FraudGNN (2-layer GraphSAGE) for MI455X / gfx1250, wave32.
// fp32 WMMA (v_wmma_f32_16x16x4_f32) for the layer-1 GEMMs; workload is
// L2/atomic-bound (all operands fit in 192MB L2), so fp32 matrix math is free.
// W1l/W1r are pre-transposed + zero-padded into workspace so the WMMA K-loop
// is branchless and B fragments load as single b64s.

typedef float v2f __attribute__((ext_vector_type(2)));
typedef float v8f __attribute__((ext_vector_type(8)));

#define N_NODES 100000
#define N_EDGES 1600000
#define F_IN 166
#define F_HID 128
#define KPAD 168  // F_IN padded to multiple of 4 (zero-filled tail rows)
#define LDK 170   // LDS row stride: 170 mod 64 banks -> conflict-free

__global__ void zero_f32(float* __restrict__ p, size_t n) {
  size_t i = (size_t)blockIdx.x * blockDim.x + threadIdx.x;
  size_t stride = (size_t)gridDim.x * blockDim.x;
  for (; i < n; i += stride) p[i] = 0.0f;
}

// Wt[n][k] = (k < 166) ? W[k][n] : 0   (transpose 166x128 -> 128x168 padded)
__global__ void transpose_pad_w(const float* __restrict__ W,
                                float* __restrict__ Wt) {
  int i = blockIdx.x * blockDim.x + threadIdx.x;
  if (i >= F_HID * KPAD) return;
  int n = i / KPAD, k = i % KPAD;
  Wt[i] = (k < F_IN) ? W[(size_t)k * F_HID + n] : 0.0f;
}

// One wave32 per edge: lanes stride the feature row, float atomics into agg.
// When count_deg != 0, lane 0 also accumulates the in-degree.
__global__ void scatter_add(const float* __restrict__ feat,
                            const long long* __restrict__ src,
                            const long long* __restrict__ dst,
                            float* __restrict__ agg, float* __restrict__ deg,
                            int n_edges, int F, int count_deg) {
  int wave = (blockIdx.x * blockDim.x + threadIdx.x) >> 5;
  int lane = threadIdx.x & 31;
  int nWaves = (gridDim.x * blockDim.x) >> 5;
  for (int e = wave; e < n_edges; e += nWaves) {
    long long s = src[e], d = dst[e];
    const float* __restrict__ row = feat + (size_t)s * F;
    float* __restrict__ out = agg + (size_t)d * F;
    if (count_deg && lane == 0) atomicAdd(&deg[(int)d], 1.0f);
    for (int f = lane; f < F; f += 32)
      atomicAdd(&out[f], row[f]);
  }
}

// h = relu( (agg1/deg) @ W1l + x @ W1r + b1 )
// Block = 256 threads = 8 waves; block computes 16 rows x 128 cols.
// Each wave: one 16x16 tile via chained v_wmma_f32_16x16x4_f32.
__global__ __launch_bounds__(256)
void gemm1_wmma(const float* __restrict__ agg,   // raw sums, N_NODES x 166
                const float* __restrict__ x,     // N_NODES x 166
                const float* __restrict__ deg,   // N_NODES
                const float* __restrict__ Wtl,   // 128 x 168 (transposed, padded)
                const float* __restrict__ Wtr,   // 128 x 168
                const float* __restrict__ b1,    // 128
                float* __restrict__ h) {         // N_NODES x 128
  __shared__ float As[16 * LDK];
  __shared__ float Xs[16 * LDK];
  __shared__ float rs[16];

  const int tid = threadIdx.x;
  const int m0 = blockIdx.x * 16;

  if (tid < 16) rs[tid] = 1.0f / fmaxf(deg[m0 + tid], 1.0f);
  __syncthreads();

  // Stage 16 normalized agg rows + 16 x rows into LDS (zero-pad K tail).
  for (int i = tid; i < 16 * LDK; i += 256) {
    int r = i / LDK, c = i % LDK;
    float a = 0.0f, xx = 0.0f;
    if (c < F_IN) {
      size_t off = (size_t)(m0 + r) * F_IN + c;
      a = agg[off] * rs[r];
      xx = x[off];
    }
    As[i] = a;
    Xs[i] = xx;
  }
  __syncthreads();

  const int wave = tid >> 5;
  const int lane = tid & 31;
  const int nn = wave * 16 + (lane & 15); // output column for this lane
  const int khalf = (lane >> 4) * 2;      // A/B: lanes 0-15 -> K+0,1; 16-31 -> K+2,3
  const int rowA = lane & 15;             // A-matrix row held by this lane

  const float* __restrict__ wl = Wtl + (size_t)nn * KPAD;
  const float* __restrict__ wr = Wtr + (size_t)nn * KPAD;

  v8f c = {};
  for (int k0 = 0; k0 < F_IN; k0 += 4) {
    const int kb = k0 + khalf;            // kb+1 <= 167 < KPAD: branchless
    v2f a, ax;
    a.x  = As[rowA * LDK + kb];
    a.y  = As[rowA * LDK + kb + 1];
    ax.x = Xs[rowA * LDK + kb];
    ax.y = Xs[rowA * LDK + kb + 1];
    v2f bl = *(const v2f*)(wl + kb);      // 8B-aligned: KPAD*4 and kb*4 mult of 8
    v2f br = *(const v2f*)(wr + kb);
    // D = A*B + C, chained fp32 accumulate
    c = __builtin_amdgcn_wmma_f32_16x16x4_f32(false, a,  false, bl, (short)0, c, false, false);
    c = __builtin_amdgcn_wmma_f32_16x16x4_f32(false, ax, false, br, (short)0, c, false, false);
  }

  // Epilogue: bias + relu + store. C/D layout: VGPR r holds M=r (lanes 0-15)
  // and M=r+8 (lanes 16-31), N = lane&15.
  const float bias = b1[nn];
  const int rbase = (lane >> 4) * 8;
  for (int r = 0; r < 8; ++r) {
    float v = c[r] + bias;
    v = fmaxf(v, 0.0f);
    h[(size_t)(m0 + rbase + r) * F_HID + nn] = v;
  }
}

// out = log_softmax( (agg2/deg) @ W2l + h @ W2r + b2 ), N=2 -> VALU kernel.
// One wave32 per node; lane-parallel dot products + shfl_xor reduction.
__global__ void final_layer(const float* __restrict__ h,     // N_NODES x 128
                            const float* __restrict__ agg2,  // raw sums
                            const float* __restrict__ deg,
                            const float* __restrict__ W2l,   // 128 x 2
                            const float* __restrict__ W2r,   // 128 x 2
                            const float* __restrict__ b2,    // 2
                            float* __restrict__ out) {       // N_NODES x 2
  int node = (blockIdx.x * blockDim.x + threadIdx.x) >> 5;
  int lane = threadIdx.x & 31;
  if (node >= N_NODES) return;
  const float rsc = 1.0f / fmaxf(deg[node], 1.0f);
  const float* __restrict__ hr = h + (size_t)node * F_HID;
  const float* __restrict__ ar = agg2 + (size_t)node * F_HID;
  float z0 = 0.0f, z1 = 0.0f;
  for (int f = lane; f < F_HID; f += 32) {
    float hv = hr[f];
    float av = ar[f] * rsc;
    z0 += hv * W2r[f * 2 + 0] + av * W2l[f * 2 + 0];
    z1 += hv * W2r[f * 2 + 1] + av * W2l[f * 2 + 1];
  }
  for (int off = 16; off > 0; off >>= 1) {
    z0 += __shfl_xor(z0, off, 32);
    z1 += __shfl_xor(z1, off, 32);
  }
  if (lane == 0) {
    z0 += b2[0];
    z1 += b2[1];
    float m = fmaxf(z0, z1);
    float l = m + logf(expf(z0 - m) + expf(z1 - m));
    out[(size_t)node * 2 + 0] = z0 - l;
    out[(size_t)node * 2 + 1] = z1 - l;
  }
}

extern "C" void kernel_launch(void* const* d_in, const int* in_sizes, int n_in,
                              void* d_out, int out_size, void* d_ws, size_t ws_size,
                              hipStream_t stream) {
  const float*     x    = (const float*)d_in[0];
  const long long* edge = (const long long*)d_in[1];  // int64 (2, N_EDGES)
  const float*     W1l  = (const float*)d_in[2];
  const float*     b1   = (const float*)d_in[3];
  const float*     W1r  = (const float*)d_in[4];
  const float*     W2l  = (const float*)d_in[5];
  const float*     b2   = (const float*)d_in[6];
  const float*     W2r  = (const float*)d_in[7];
  float* out = (float*)d_out;

  const long long* src = edge;            // edge_index[0]
  const long long* dst = edge + N_EDGES;  // edge_index[1]

  // Workspace (floats): [deg | agg (166-wide max) | h | Wt_l | Wt_r]
  float* ws_f = (float*)d_ws;
  float* deg  = ws_f;
  float* agg  = ws_f + N_NODES;
  float* hbuf = agg + (size_t)N_NODES * F_IN;
  float* Wtl  = hbuf + (size_t)N_NODES * F_HID;
  float* Wtr  = Wtl + (size_t)F_HID * KPAD;

  // Prep: transpose+pad weights, zero deg+agg1.
  transpose_pad_w<<<(F_HID * KPAD + 255) / 256, 256, 0, stream>>>(W1l, Wtl);
  transpose_pad_w<<<(F_HID * KPAD + 255) / 256, 256, 0, stream>>>(W1r, Wtr);
  zero_f32<<<2048, 256, 0, stream>>>(ws_f, (size_t)N_NODES + (size_t)N_NODES * F_IN);

  // ---- Layer 1 ----
  scatter_add<<<(N_EDGES + 7) / 8, 256, 0, stream>>>(x, src, dst, agg, deg,
                                                     N_EDGES, F_IN, 1);
  gemm1_wmma<<<N_NODES / 16, 256, 0, stream>>>(agg, x, deg, Wtl, Wtr, b1, hbuf);

  // ---- Layer 2 (reuse agg buffer for 128-wide sums) ----
  zero_f32<<<2048, 256, 0, stream>>>(agg, (size_t)N_NODES * F_HID);
  scatter_add<<<(N_EDGES + 7) / 8, 256, 0, stream>>>(hbuf, src, dst, agg, deg,
                                                     N_EDGES, F_HID, 0);
  final_layer<<<((size_t)N_NODES * 32 + 255) / 256, 256, 0, stream>>>(
      hbuf, agg, deg, W2l, W2r, b2, out);
}